// EncoderClass_9285719294035
// MI455X (gfx1250) — compile-verified
//
#include <hip/hip_runtime.h>

typedef float v2f __attribute__((ext_vector_type(2)));
typedef float v8f __attribute__((ext_vector_type(8)));

#define ND_   50000
#define NG_   100000
#define E_    600000
#define DIN_  128
#define H_    64

// ---------------------------------------------------------------------------
// Edge-count kernel: cnt[dst[e]] += 1.0f  (counts reused by both layers)
// ---------------------------------------------------------------------------
__global__ __launch_bounds__(256)
void count_edges(const int* __restrict__ dst, float* __restrict__ cnt, int nE) {
    int i = blockIdx.x * blockDim.x + threadIdx.x;
    if (i < nE) atomicAdd(&cnt[dst[i]], 1.0f);
}

// ---------------------------------------------------------------------------
// Scatter-add: acc[dst[e]*D + d] += xsrc[src[e]*D + d], D = 1<<LOGD.
// One float4 per thread (coalesced within an edge); the accumulation buffers
// (<=51MB) are L2-resident on MI455X (192MB L2), so the f32 atomics resolve
// at the L2 atomic units rather than HBM.
// ---------------------------------------------------------------------------
template<int LOGD>
__global__ __launch_bounds__(256)
void scatter_add4(const int* __restrict__ src, const int* __restrict__ dst,
                  const float* __restrict__ xsrc, float* __restrict__ acc, int nE) {
    const unsigned CPT = 1u << (LOGD - 2);      // float4 chunks per edge
    unsigned idx = blockIdx.x * blockDim.x + threadIdx.x;
    unsigned e   = idx >> (LOGD - 2);
    if (e >= (unsigned)nE) return;
    unsigned c = idx & (CPT - 1);

    const float4* xs = (const float4*)(xsrc + ((size_t)src[e] << LOGD));
    float4 v = xs[c];
    float* out = acc + ((size_t)dst[e] << LOGD) + (size_t)c * 4;
    atomicAdd(out + 0, v.x);
    atomicAdd(out + 1, v.y);
    atomicAdd(out + 2, v.z);
    atomicAdd(out + 3, v.w);
}

// ---------------------------------------------------------------------------
// Fused SAGE layer:  out = (msum/max(cnt,1)) @ Wl + bl + xdst @ Wr
// K = input dim (128 layer1 / 64 layer2); output dim fixed at 64.
//
// Block = 128 threads (4 waves). Each wave owns TWO 16-row tiles (32 rows x
// all 64 output cols, 8 x v8f accumulators) so every B fragment feeds two
// v_wmma_f32_16x16x4_f32 ops (2x LDS-traffic reuse).
//
// Weights are staged in LDS pre-packed as K-pairs:
//   Wpk[pp][col] = { W[2pp][col], W[2pp+1][col] },  pp in [0, K)
// (Wl -> pairs [0, K/2), Wr -> pairs [K/2, K)), so the per-lane B fragment
// (rows k+khalf, k+khalf+1; khalf = 2*(lane>>4)) is ONE contiguous
// ds_load_b64 straight into an even-aligned VGPR pair — no repack movs.
// A fragments are contiguous float2 from global (global_load_b64), scaled by
// the folded 1/max(cnt,1).
//
// N % 16 == 0 for both node sets, so 16-row tiles are always full; the odd
// tile count of ND (3125) is handled by a wave-uniform tail (clamped A rows,
// uniform store guard) keeping EXEC all-ones as WMMA requires.
// ---------------------------------------------------------------------------
template<int K>
__global__ __launch_bounds__(128)
void sage_wmma(const float* __restrict__ msum, const float* __restrict__ cnt,
               const float* __restrict__ xdst,
               const float* __restrict__ Wl, const float* __restrict__ bl,
               const float* __restrict__ Wr,
               float* __restrict__ out, int N)
{
    __shared__ v2f Wpk[K * 64];                 // K pairs x 64 cols, 8B each
    {
        float* lds = (float*)Wpk;
        for (int i = threadIdx.x; i < K * 64; i += blockDim.x) {
            const int k = i >> 6, col = i & 63;
            lds[((k >> 1) * 64 + col) * 2 + (k & 1)]           = Wl[i];
            lds[((K / 2 + (k >> 1)) * 64 + col) * 2 + (k & 1)] = Wr[i];
        }
    }
    __syncthreads();

    const int wave = threadIdx.x >> 5;
    const int lane = threadIdx.x & 31;
    const int pair = blockIdx.x * 4 + wave;     // 32-row tile-pair index
    const int row0 = pair * 32;
    if (row0 >= N) return;                      // wave-uniform; no barriers after
    const bool has2 = (row0 + 32 <= N);         // wave-uniform tail flag

    const int m  = lane & 15;                   // A row within tile / B column
    const int ph = lane >> 4;                   // K-pair offset for this half-wave
    const int rowA0 = row0 + m;
    const int rowA1 = min(row0 + 16 + m, N - 1);   // clamp: loads valid, stores guarded

    const float s0 = 1.0f / fmaxf(cnt[rowA0], 1.0f);
    const float s1 = 1.0f / fmaxf(cnt[rowA1], 1.0f);

    v8f acc0[4] = {};
    v8f acc1[4] = {};

    // ---- mean @ Wl ----
    {
        const v2f* A0 = (const v2f*)(msum + (size_t)rowA0 * K);
        const v2f* A1 = (const v2f*)(msum + (size_t)rowA1 * K);
        for (int k = 0; k < K; k += 4) {
            const int kp = (k >> 1) + ph;       // pair idx in Wl section & in A row
            v2f a0 = A0[kp] * s0;
            v2f a1 = A1[kp] * s1;
#pragma unroll
            for (int t = 0; t < 4; ++t) {
                v2f b = Wpk[kp * 64 + t * 16 + m];
                acc0[t] = __builtin_amdgcn_wmma_f32_16x16x4_f32(
                    false, a0, false, b, (short)0, acc0[t], false, false);
                acc1[t] = __builtin_amdgcn_wmma_f32_16x16x4_f32(
                    false, a1, false, b, (short)0, acc1[t], false, false);
            }
        }
    }

    // ---- + xdst @ Wr ----
    {
        const v2f* X0 = (const v2f*)(xdst + (size_t)rowA0 * K);
        const v2f* X1 = (const v2f*)(xdst + (size_t)rowA1 * K);
        for (int k = 0; k < K; k += 4) {
            const int kp = (k >> 1) + ph;
            v2f a0 = X0[kp];
            v2f a1 = X1[kp];
#pragma unroll
            for (int t = 0; t < 4; ++t) {
                v2f b = Wpk[(K / 2 + kp) * 64 + t * 16 + m];
                acc0[t] = __builtin_amdgcn_wmma_f32_16x16x4_f32(
                    false, a0, false, b, (short)0, acc0[t], false, false);
                acc1[t] = __builtin_amdgcn_wmma_f32_16x16x4_f32(
                    false, a1, false, b, (short)0, acc1[t], false, false);
            }
        }
    }

    // ---- store + bias; C/D layout: VGPR r -> M=r (lanes 0-15) / M=r+8 (16-31)
    const int colb  = lane & 15;
    const int mb0   = row0 + 8 * ph;
#pragma unroll
    for (int t = 0; t < 4; ++t) {
        const float bias = bl[t * 16 + colb];
#pragma unroll
        for (int r = 0; r < 8; ++r)
            out[(size_t)(mb0 + r) * 64 + t * 16 + colb] = acc0[t][r] + bias;
        if (has2) {
#pragma unroll
            for (int r = 0; r < 8; ++r)
                out[(size_t)(mb0 + 16 + r) * 64 + t * 16 + colb] = acc1[t][r] + bias;
        }
    }
}

// ---------------------------------------------------------------------------
// Host launcher
// ---------------------------------------------------------------------------
extern "C" void kernel_launch(void* const* d_in, const int* in_sizes, int n_in,
                              void* d_out, int out_size, void* d_ws, size_t ws_size,
                              hipStream_t stream) {
    const float* x_disease = (const float*)d_in[0];
    const float* x_gene    = (const float*)d_in[1];
    const int*   src_dg    = (const int*)d_in[2];
    const int*   dst_dg    = (const int*)d_in[3];
    const int*   src_gd    = (const int*)d_in[4];
    const int*   dst_gd    = (const int*)d_in[5];
    const float* Wl1_dg = (const float*)d_in[6];
    const float* bl1_dg = (const float*)d_in[7];
    const float* Wr1_dg = (const float*)d_in[8];
    const float* Wl1_gd = (const float*)d_in[9];
    const float* bl1_gd = (const float*)d_in[10];
    const float* Wr1_gd = (const float*)d_in[11];
    const float* Wl2_dg = (const float*)d_in[12];
    const float* bl2_dg = (const float*)d_in[13];
    const float* Wr2_dg = (const float*)d_in[14];
    const float* Wl2_gd = (const float*)d_in[15];
    const float* bl2_gd = (const float*)d_in[16];
    const float* Wr2_gd = (const float*)d_in[17];

    // Output: d2 [ND,64] then g2 [NG,64]
    float* d2 = (float*)d_out;
    float* g2 = d2 + (size_t)ND_ * 64;

    // Workspace carve-up (floats). msg-sum buffers are reused (128-wide in
    // layer 1, 64-wide alias in layer 2). Total ~116 MB.
    float* ws = (float*)d_ws;
    size_t off = 0;
    float* ms_g  = ws + off; off += (size_t)NG_ * DIN_;   // segment sums -> genes
    float* ms_d  = ws + off; off += (size_t)ND_ * DIN_;   // segment sums -> diseases
    float* cnt_g = ws + off; off += (size_t)NG_;
    float* cnt_d = ws + off; off += (size_t)ND_;
    float* g1    = ws + off; off += (size_t)NG_ * H_;
    float* d1    = ws + off; off += (size_t)ND_ * H_;
    (void)off; (void)ws_size; (void)in_sizes; (void)n_in; (void)out_size;

    const int TPB = 256;
    const int eb  = (E_ + TPB - 1) / TPB;

    // ---- zero accumulators (memset nodes are graph-capture legal) ----
    hipMemsetAsync(ms_g,  0, (size_t)NG_ * DIN_ * sizeof(float), stream);
    hipMemsetAsync(ms_d,  0, (size_t)ND_ * DIN_ * sizeof(float), stream);
    hipMemsetAsync(cnt_g, 0, (size_t)NG_ * sizeof(float), stream);
    hipMemsetAsync(cnt_d, 0, (size_t)ND_ * sizeof(float), stream);

    // ---- edge degree counts (shared by both layers) ----
    count_edges<<<eb, TPB, 0, stream>>>(dst_dg, cnt_g, E_);
    count_edges<<<eb, TPB, 0, stream>>>(dst_gd, cnt_d, E_);

    // ---- layer 1 segment sums (128-wide) ----
    {
        const int blocks = ((E_ << (7 - 2)) + TPB - 1) / TPB;   // E * 32 chunk-threads
        scatter_add4<7><<<blocks, TPB, 0, stream>>>(src_dg, dst_dg, x_disease, ms_g, E_);
        scatter_add4<7><<<blocks, TPB, 0, stream>>>(src_gd, dst_gd, x_gene,    ms_d, E_);
    }

    // ---- layer 1 fused SAGE GEMMs (K=128) ----
    {
        const int pairs_g = (NG_ / 16 + 1) / 2;   // 32-row tile pairs
        const int pairs_d = (ND_ / 16 + 1) / 2;
        sage_wmma<DIN_><<<(pairs_g + 3) / 4, 128, 0, stream>>>(
            ms_g, cnt_g, x_gene,    Wl1_dg, bl1_dg, Wr1_dg, g1, NG_);
        sage_wmma<DIN_><<<(pairs_d + 3) / 4, 128, 0, stream>>>(
            ms_d, cnt_d, x_disease, Wl1_gd, bl1_gd, Wr1_gd, d1, ND_);
    }

    // ---- layer 2 segment sums (64-wide, reuse ms_* buffers) ----
    hipMemsetAsync(ms_g, 0, (size_t)NG_ * H_ * sizeof(float), stream);
    hipMemsetAsync(ms_d, 0, (size_t)ND_ * H_ * sizeof(float), stream);
    {
        const int blocks = ((E_ << (6 - 2)) + TPB - 1) / TPB;   // E * 16 chunk-threads
        scatter_add4<6><<<blocks, TPB, 0, stream>>>(src_dg, dst_dg, d1, ms_g, E_);
        scatter_add4<6><<<blocks, TPB, 0, stream>>>(src_gd, dst_gd, g1, ms_d, E_);
    }

    // ---- layer 2 fused SAGE GEMMs (K=64), write straight to d_out ----
    {
        const int pairs_g = (NG_ / 16 + 1) / 2;
        const int pairs_d = (ND_ / 16 + 1) / 2;
        sage_wmma<H_><<<(pairs_g + 3) / 4, 128, 0, stream>>>(
            ms_g, cnt_g, g1, Wl2_dg, bl2_dg, Wr2_dg, g2, NG_);
        sage_wmma<H_><<<(pairs_d + 3) / 4, 128, 0, stream>>>(
            ms_d, cnt_d, d1, Wl2_gd, bl2_gd, Wr2_gd, d2, ND_);
    }
}